// LightGCN_59244778881391
// MI455X (gfx1250) — compile-verified
//
#include <hip/hip_runtime.h>
#include <hip/hip_bf16.h>

typedef __attribute__((ext_vector_type(2))) float v2f;
typedef __attribute__((ext_vector_type(8))) float v8f;

#define WAVE 32
#define EMBD 64

// ---------------- helpers ----------------

__global__ void k_zero_i32(int* __restrict__ p, int n) {
  int i = blockIdx.x * blockDim.x + threadIdx.x;
  if (i < n) p[i] = 0;
}

// acc = cur = concat(user_emb, item_emb), float4-vectorized (both tables are
// 64-dim rows -> element counts divisible by 4)
__global__ void k_init_emb(const float4* __restrict__ ue4, const float4* __restrict__ ie4,
                           float4* __restrict__ acc4, float4* __restrict__ cur4,
                           int uQuads, int totQuads) {
  int i = blockIdx.x * blockDim.x + threadIdx.x;
  if (i >= totQuads) return;
  float4 v = (i < uQuads) ? ue4[i] : ie4[i - uQuads];
  acc4[i] = v;
  cur4[i] = v;
}

// per-row degree histogram (int atomics, cheap: 4M ops once)
__global__ void k_hist(const int* __restrict__ rows, int* __restrict__ cnt, int nnz) {
  int e = blockIdx.x * blockDim.x + threadIdx.x;
  if (e < nnz) atomicAdd(&cnt[rows[e]], 1);
}

// single-workgroup exclusive scan over N row counts -> rowPtr[N+1], rowFill copy
__global__ void k_scan(const int* __restrict__ cnt, int* __restrict__ rowPtr,
                       int* __restrict__ rowFill, int n) {
  __shared__ int sd[1024];
  __shared__ int carry;
  int t = threadIdx.x;
  if (t == 0) carry = 0;
  __syncthreads();
  for (int base = 0; base < n; base += 1024) {
    int x = (base + t < n) ? cnt[base + t] : 0;
    sd[t] = x;
    __syncthreads();
    for (int off = 1; off < 1024; off <<= 1) {
      int v = (t >= off) ? sd[t - off] : 0;
      __syncthreads();
      sd[t] += v;
      __syncthreads();
    }
    int incl = sd[t];
    int excl = carry + incl - x;
    if (base + t < n) { rowPtr[base + t] = excl; rowFill[base + t] = excl; }
    __syncthreads();
    if (t == 1023) carry += incl;
    __syncthreads();
  }
  if (t == 0) rowPtr[n] = carry;
}

// CSR fill: place (col,val) into row buckets
__global__ void k_fill(const int* __restrict__ rows, const int* __restrict__ cols,
                       const float* __restrict__ vals, int* __restrict__ rowFill,
                       int* __restrict__ colS, float* __restrict__ valS, int nnz) {
  int e = blockIdx.x * blockDim.x + threadIdx.x;
  if (e >= nnz) return;
  int r = rows[e];
  int p = atomicAdd(&rowFill[r], 1);
  colS[p] = cols[e];
  valS[p] = vals[e];
}

// Atomic-free pull SpMM: one wave32 per row, lane owns dims (2*lane, 2*lane+1)
// so each neighbor gather is a single aligned global_load_b64.  Neighbor
// (col,val) are broadcast with v_readlane (uniform lane index) -> scalar row
// base address, scalar scale factor; no LDS permute traffic in the hot loop.
// Fuses the layer accumulation acc += next.  All gathers hit the 192MB L2
// (table is 38.4MB).
__global__ void __launch_bounds__(256) k_spmm(
    const int* __restrict__ rowPtr, const int* __restrict__ colS,
    const float* __restrict__ valS, const float* __restrict__ cur,
    float* __restrict__ nxt, float* __restrict__ acc, int n) {
  int lane = threadIdx.x & (WAVE - 1);
  int row = blockIdx.x * (blockDim.x / WAVE) + (threadIdx.x / WAVE);
  if (row >= n) return;
  int beg = rowPtr[row];
  int end = rowPtr[row + 1];
  float s0 = 0.f, s1 = 0.f;
  for (int e0 = beg; e0 < end; e0 += WAVE) {
    __builtin_prefetch(&colS[e0 + WAVE + lane], 0, 1);   // global_prefetch_b8
    __builtin_prefetch(&valS[e0 + WAVE + lane], 0, 1);
    int idx = e0 + lane;
    int c = 0; float v = 0.f;
    if (idx < end) { c = colS[idx]; v = valS[idx]; }
    int cnt = min(WAVE, end - e0);
    for (int j = 0; j < cnt; ++j) {
      int   cc = __builtin_amdgcn_readlane(c, j);                       // uniform
      float vv = __int_as_float(__builtin_amdgcn_readlane(__float_as_int(v), j));
      const float2* rp = (const float2*)(cur + (size_t)cc * EMBD);
      float2 e = rp[lane];                                              // b64 gather
      s0 += vv * e.x;
      s1 += vv * e.y;
    }
  }
  size_t o = (size_t)row * EMBD + 2 * lane;
  *(float2*)(nxt + o) = make_float2(s0, s1);
  float2 a = *(const float2*)(acc + o);
  a.x += s0; a.y += s1;
  *(float2*)(acc + o) = a;
}

// Batched dot interactions via V_WMMA_F32_16X16X4_F32.
// One wave per 16-element batch tile; 16 K=4 WMMA steps accumulate
// Fu(16x64) x Fi^T(64x16) in fp32; diagonal extracted per C/D layout.
// A-layout (16x4 f32): lane m=l&15, kh=l>>4 -> VGPR0=K(2kh), VGPR1=K(2kh+1).
// B-layout (4x16 f32): lanes 0-15 hold K=0,1; lanes 16-31 hold K=2,3 -> same
// per-lane offsets as A since B = Fi^T with N = l&15.
__global__ void __launch_bounds__(256) k_final(
    const float* __restrict__ acc, const int* __restrict__ uids,
    const int* __restrict__ iids, float* __restrict__ out,
    int U, int batch, int ntiles) {
  int lane = threadIdx.x & (WAVE - 1);
  int wv = blockIdx.x * (blockDim.x / WAVE) + (threadIdx.x / WAVE);
  if (wv >= ntiles) return;                 // wave-uniform: EXEC stays all-ones
  int m  = lane & 15;
  int kh = lane >> 4;
  int bi = wv * 16 + m;
  int bidx = min(bi, batch - 1);
  const float* ru = acc + (size_t)uids[bidx] * EMBD;
  const float* ri = acc + ((size_t)U + iids[bidx]) * EMBD;
  v8f c = {};
#pragma unroll
  for (int s = 0; s < 16; ++s) {
    int o = 4 * s + 2 * kh;
    v2f a, b;
    a.x = ru[o]; a.y = ru[o + 1];
    b.x = ri[o]; b.y = ri[o + 1];
    c = __builtin_amdgcn_wmma_f32_16x16x4_f32(false, a, false, b,
                                              (short)0, c, false, false);
  }
  // D[j + 8*kh][m] lives in c[j]; diagonal needs j = m - 8*kh in [0,8)
  int j = m - 8 * kh;
  float r = 0.f;
#pragma unroll
  for (int jj = 0; jj < 8; ++jj)
    if (jj == j) r = c[jj];
  bool valid = (kh == 0) ? (m < 8) : (m >= 8);
  if (valid && bi < batch) out[bi] = r * (1.0f / 16.0f);  // (acc/4)·(acc/4)
}

// ---------------- launch ----------------

extern "C" void kernel_launch(void* const* d_in, const int* in_sizes, int n_in,
                              void* d_out, int out_size, void* d_ws, size_t ws_size,
                              hipStream_t stream) {
  const float* ue   = (const float*)d_in[0];
  const float* ie   = (const float*)d_in[1];
  const float* vals = (const float*)d_in[2];
  const int*   rows = (const int*)d_in[3];
  const int*   cols = (const int*)d_in[4];
  const int*   uids = (const int*)d_in[5];
  const int*   iids = (const int*)d_in[6];
  float* out = (float*)d_out;

  int uElems = in_sizes[0];
  int iElems = in_sizes[1];
  int nnz    = in_sizes[2];
  int batch  = in_sizes[5];
  int U  = uElems / EMBD;
  int N  = U + iElems / EMBD;
  int ND = N * EMBD;

  char* base = (char*)d_ws;
  size_t off = 0;
  auto alloc = [&](size_t bytes) -> void* {
    void* p = base + off;
    off += bytes;
    off = (off + 255) & ~(size_t)255;
    return p;
  };
  float* acc     = (float*)alloc((size_t)ND * 4);
  float* bufA    = (float*)alloc((size_t)ND * 4);
  float* bufB    = (float*)alloc((size_t)ND * 4);
  int*   rowCnt  = (int*)alloc((size_t)N * 4);
  int*   rowPtr  = (int*)alloc((size_t)(N + 1) * 4);
  int*   rowFill = (int*)alloc((size_t)N * 4);
  int*   colS    = (int*)alloc((size_t)nnz * 4);
  float* valS    = (float*)alloc((size_t)nnz * 4);
  (void)ws_size; (void)n_in; (void)out_size;

  k_zero_i32<<<(N + 255) / 256, 256, 0, stream>>>(rowCnt, N);
  int uQuads = uElems / 4, totQuads = ND / 4;
  k_init_emb<<<(totQuads + 255) / 256, 256, 0, stream>>>(
      (const float4*)ue, (const float4*)ie, (float4*)acc, (float4*)bufA,
      uQuads, totQuads);
  k_hist<<<(nnz + 255) / 256, 256, 0, stream>>>(rows, rowCnt, nnz);
  k_scan<<<1, 1024, 0, stream>>>(rowCnt, rowPtr, rowFill, N);
  k_fill<<<(nnz + 255) / 256, 256, 0, stream>>>(rows, cols, vals, rowFill, colS, valS, nnz);

  int spmmBlocks = (N + 7) / 8;  // 8 waves per 256-thread block, 1 wave/row
  float* cur = bufA;
  float* nxt = bufB;
  for (int l = 0; l < 3; ++l) {
    k_spmm<<<spmmBlocks, 256, 0, stream>>>(rowPtr, colS, valS, cur, nxt, acc, N);
    float* t = cur; cur = nxt; nxt = t;
  }

  int ntiles = (batch + 15) / 16;
  k_final<<<(ntiles + 7) / 8, 256, 0, stream>>>(acc, uids, iids, out, U, batch, ntiles);
}